// RNNEncoder_34033320853743
// MI455X (gfx1250) — compile-verified
//
#include <hip/hip_runtime.h>
#include <cstddef>

// Problem constants (match reference)
#define RNN_BATCH  256
#define RNN_SEQ    1024
#define RNN_EMBED  1024
#define RNN_HIDDEN 1024

// GEMM tiling: 256 threads = 8 wave32s, 64x64 output tile, K-tile 32.
#define BM 64
#define BN 64
#define BK 32
#define LDA 36                           // padded LDS row stride (floats):
                                         //  - 36r mod 64 distinct for r=0..15 -> conflict-free
                                         //  - row*36 + 4k offsets are 16B aligned (b128 stores)
                                         //  - row*36 + even k is 8B aligned (b64 frag reads)
#define KTOT (RNN_EMBED + RNN_HIDDEN)    // fused [ex_t , h] @ [Wi ; Wh]^T
#define NIT  (KTOT / BK)                 // 64 K-tiles

typedef __attribute__((ext_vector_type(2))) float v2f;
typedef __attribute__((ext_vector_type(8))) float v8f;
typedef __attribute__((ext_vector_type(4))) int   v4i;

// ---- CDNA5 async global->LDS copy path (ASYNCcnt), with safe fallback ----
#if __has_builtin(__builtin_amdgcn_global_load_async_to_lds_b128)
#define RNN_ASYNC 1
typedef __attribute__((address_space(1))) v4i* rnn_g4p;   // global int4*
typedef __attribute__((address_space(3))) v4i* rnn_l4p;   // LDS int4*
__device__ __forceinline__ void rnn_async_cp16(const float* g, float* l) {
    __builtin_amdgcn_global_load_async_to_lds_b128((rnn_g4p)g, (rnn_l4p)l, 0, 0);
}
__device__ __forceinline__ void rnn_wait_async_hot() {   // retire older tile, keep 4 in flight
#if __has_builtin(__builtin_amdgcn_s_wait_asynccnt)
    __builtin_amdgcn_s_wait_asynccnt(4);
#else
    asm volatile("s_wait_asynccnt 0x4" ::: "memory");
#endif
}
__device__ __forceinline__ void rnn_wait_async_all() {
#if __has_builtin(__builtin_amdgcn_s_wait_asynccnt)
    __builtin_amdgcn_s_wait_asynccnt(0);
#else
    asm volatile("s_wait_asynccnt 0x0" ::: "memory");
#endif
}
#else
#define RNN_ASYNC 0
#endif

// bc = bi + bh (biases fuse: tanh(ex*Wi^T + h*Wh^T + bi + bh))
__global__ __launch_bounds__(256) void rnn_bias_combine(const float* __restrict__ bi,
                                                        const float* __restrict__ bh,
                                                        float* __restrict__ bc) {
    int i = blockIdx.x * blockDim.x + threadIdx.x;
    if (i < RNN_HIDDEN) bc[i] = bi[i] + bh[i];
}

__global__ __launch_bounds__(256) void rnn_zero(float* __restrict__ p, int n) {
    int i = blockIdx.x * blockDim.x + threadIdx.x;
    if (i < n) p[i] = 0.0f;
}

// One recurrent step, fused with embedding gather + input projection:
//   h_out[b,h] = tanh( sum_e emb[x[b,t],e]*Wi[h,e] + sum_k h_in[b,k]*Wh[h,k] + bc[h] )
// M=256 (batch), N=1024 (hidden), K=2048 (1024 emb phase + 1024 hidden phase).
__global__ __launch_bounds__(256) void rnn_step(const int*   __restrict__ x,
                                                const float* __restrict__ emb,
                                                const float* __restrict__ Wi,
                                                const float* __restrict__ Wh,
                                                const float* __restrict__ bc,
                                                const float* __restrict__ h_in,
                                                float*       __restrict__ h_out,
                                                int t) {
    // Triple-buffered tiles, both stored row-major [row][k] with padded stride.
    __shared__ __align__(16) float As[3][BM][LDA];   // A rows (gathered emb / h)
    __shared__ __align__(16) float Bs[3][BN][LDA];   // weight rows (Wi / Wh), no transpose

    const int tid  = threadIdx.x;
    const int lane = tid & 31;          // wave32
    const int wv   = tid >> 5;          // 0..7
    const int wm   = wv >> 1;           // 0..3 : wave row  (16 rows each)
    const int wn   = wv & 1;            // 0..1 : wave col  (32 cols each)
    const int half = lane >> 4;         // 0/1  : lane half
    const int l16  = lane & 15;

    const int b0 = blockIdx.x * BM;     // batch tile
    const int n0 = blockIdx.y * BN;     // hidden tile

    // Cooperative loaders: thread -> (row, two 16B chunks at cq and cq+16)
    const int rowL = tid >> 2;          // 0..63
    const int cq   = (tid & 3) << 2;    // 0,4,8,12

    const int bA  = b0 + rowL;
    const int tok = x[bA * RNN_SEQ + t];
    const float* emb_row = emb + (size_t)tok * RNN_EMBED + cq;
    const float* h_row   = h_in + (size_t)bA * RNN_HIDDEN + cq;
    const float* wi_row  = Wi + (size_t)(n0 + rowL) * RNN_EMBED + cq;
    const float* wh_row  = Wh + (size_t)(n0 + rowL) * RNN_HIDDEN + cq;

    auto asrc = [&](int it) -> const float* {
        const int k0 = it * BK;
        return (k0 < RNN_EMBED) ? (emb_row + k0) : (h_row + (k0 - RNN_EMBED));
    };
    auto bsrc = [&](int it) -> const float* {
        const int k0 = it * BK;
        return (k0 < RNN_EMBED) ? (wi_row + k0) : (wh_row + (k0 - RNN_EMBED));
    };

#if RNN_ASYNC
    auto issue = [&](int it, int buf) {   // 4 async b128 copies per thread per tile
        const float* a = asrc(it);
        const float* b = bsrc(it);
        rnn_async_cp16(a,      &As[buf][rowL][cq]);
        rnn_async_cp16(a + 16, &As[buf][rowL][cq + 16]);
        rnn_async_cp16(b,      &Bs[buf][rowL][cq]);
        rnn_async_cp16(b + 16, &Bs[buf][rowL][cq + 16]);
    };
    issue(0, 0);
#else
    float4 ra0, ra1, rb0, rb1;
    auto fetch = [&](int it) {
        const float* a = asrc(it);
        const float* b = bsrc(it);
        ra0 = *(const float4*)(a);      ra1 = *(const float4*)(a + 16);
        rb0 = *(const float4*)(b);      rb1 = *(const float4*)(b + 16);
    };
    auto commit = [&](int buf) {        // 16B-aligned b128 LDS stores
        *(float4*)&As[buf][rowL][cq]      = ra0;
        *(float4*)&As[buf][rowL][cq + 16] = ra1;
        *(float4*)&Bs[buf][rowL][cq]      = rb0;
        *(float4*)&Bs[buf][rowL][cq + 16] = rb1;
    };
    fetch(0);
    commit(0);
#endif

    v8f acc0 = {};   // 16x16 tile, cols [n0+wn*32,    +16)
    v8f acc1 = {};   // 16x16 tile, cols [n0+wn*32+16, +16)

    int cur = 0;
    for (int it = 0; it < NIT; ++it) {
        const int  nxt  = (cur + 1 == 3) ? 0 : cur + 1;
        const bool more = (it + 1 < NIT);

#if RNN_ASYNC
        if (more) { issue(it + 1, nxt); rnn_wait_async_hot(); }
        else      { rnn_wait_async_all(); }
        __syncthreads();    // all threads' tile-it writes visible; 3 buffers => WAR-safe
#else
        if (more) fetch(it + 1);
        __syncthreads();    // commits of tile it (from previous iteration) visible
#endif

        // 8 k-steps of V_WMMA_F32_16X16X4_F32; fragments are aligned b64 LDS loads.
        // A: k = reg + 2*half, m = l16 ; B: k = reg + 2*half, n = l16 (ISA layout)
        const float* arow = &As[cur][wm * 16 + l16][0];
        const float* br0  = &Bs[cur][wn * 32 + l16][0];
        const float* br1  = &Bs[cur][wn * 32 + 16 + l16][0];
        #pragma unroll
        for (int kk = 0; kk < BK; kk += 4) {
            const int ks = kk + 2 * half;
            v2f a   = *(const v2f*)(arow + ks);
            v2f bl  = *(const v2f*)(br0 + ks);
            v2f bh2 = *(const v2f*)(br1 + ks);
            acc0 = __builtin_amdgcn_wmma_f32_16x16x4_f32(
                false, a, false, bl, (short)0, acc0, false, false);
            acc1 = __builtin_amdgcn_wmma_f32_16x16x4_f32(
                false, a, false, bh2, (short)0, acc1, false, false);
        }

#if !RNN_ASYNC
        if (more) commit(nxt);  // writes buf != any buffer being read this iteration
#endif
        cur = nxt;
    }

    // Epilogue: C layout m = reg + 8*half, n = l16. Add combined bias, tanh.
    #pragma unroll
    for (int r = 0; r < 8; ++r) {
        const int bb = b0 + wm * 16 + r + 8 * half;
        const int hh = n0 + wn * 32 + l16;
        h_out[(size_t)bb * RNN_HIDDEN + hh]      = tanhf(acc0[r] + bc[hh]);
        h_out[(size_t)bb * RNN_HIDDEN + hh + 16] = tanhf(acc1[r] + bc[hh + 16]);
    }
}

extern "C" void kernel_launch(void* const* d_in, const int* in_sizes, int n_in,
                              void* d_out, int out_size, void* d_ws, size_t ws_size,
                              hipStream_t stream) {
    const int*   x   = (const int*)d_in[0];
    const float* emb = (const float*)d_in[1];
    const float* Wi  = (const float*)d_in[2];
    const float* bi  = (const float*)d_in[3];
    const float* Wh  = (const float*)d_in[4];
    const float* bh  = (const float*)d_in[5];
    float* out = (float*)d_out;

    // Workspace layout (~2.1 MB): combined bias + ping-pong h buffers.
    char*  ws = (char*)d_ws;
    float* bc = (float*)ws;                                   // 4 KB
    float* hA = (float*)(ws + 8192);                          // 1 MB
    float* hB = (float*)(ws + 8192 + (size_t)RNN_BATCH * RNN_HIDDEN * sizeof(float));
    float* hbuf[2] = {hA, hB};

    rnn_bias_combine<<<(RNN_HIDDEN + 255) / 256, 256, 0, stream>>>(bi, bh, bc);
    rnn_zero<<<(RNN_BATCH * RNN_HIDDEN) / 256, 256, 0, stream>>>(hA, RNN_BATCH * RNN_HIDDEN);

    const dim3 grid(RNN_BATCH / BM, RNN_HIDDEN / BN);  // 4 x 16 blocks
    for (int t = 0; t < RNN_SEQ; ++t) {
        const float* hin = hbuf[t & 1];
        float* hout = (t == RNN_SEQ - 1) ? out : hbuf[(t + 1) & 1];
        rnn_step<<<grid, 256, 0, stream>>>(x, emb, Wi, Wh, bc, hin, hout, t);
    }
}